// MultiHeadEdgeAttention_13881334300926
// MI455X (gfx1250) — compile-verified
//
#include <hip/hip_runtime.h>
#include <hip/hip_bf16.h>

#define N_NODES 100000
#define N_EDGES 1600000

typedef __attribute__((ext_vector_type(2))) float v2f;
typedef __attribute__((ext_vector_type(8))) float v8f;

// ws layout (float units): [0,4096) Wc = Wv@Wo ; [4096,4160) bc = bv@Wo+bo ;
// [8192, 8192 + N_NODES*64) y = x@Wc + bc

__global__ void prep_weights(const float* __restrict__ Wv, const float* __restrict__ bv,
                             const float* __restrict__ Wo, const float* __restrict__ bo,
                             float* __restrict__ Wc, float* __restrict__ bc) {
    int t = threadIdx.x;                       // single block, 256 threads
    for (int i = 0; i < 16; ++i) {
        int idx = t + 256 * i;                 // coalesced over the 4096 outputs
        int r = idx >> 6, c = idx & 63;
        float s = 0.f;
        for (int k = 0; k < 64; ++k) s = fmaf(Wv[r * 64 + k], Wo[k * 64 + c], s);
        Wc[idx] = s;
    }
    if (t < 64) {
        float s = bo[t];
        for (int k = 0; k < 64; ++k) s = fmaf(bv[k], Wo[k * 64 + t], s);
        bc[t] = s;
    }
}

__global__ void zero_out_kernel(float4* __restrict__ out, int n4) {
    int i = blockIdx.x * blockDim.x + threadIdx.x;
    if (i < n4) out[i] = make_float4(0.f, 0.f, 0.f, 0.f);
}

// y[n][:] = x[n][:] @ Wc + bc, fp32 via V_WMMA_F32_16X16X4_F32.
// 256 threads = 8 wave32; each wave owns a 16-row slab (block covers 128 rows).
// W is staged TRANSPOSED in LDS with row stride 68 so each lane's B-operand
// pair (W[ka][n], W[ka+1][n]) is one aligned ds_load_b64 into an even VGPR
// pair (no re-packing movs), and the 32 lanes touch 32 distinct bank pairs.
#define WT_STRIDE 68

__global__ __launch_bounds__(256) void gemm_wmma(const float* __restrict__ x,
                                                 const float* __restrict__ Wc,
                                                 const float* __restrict__ bc,
                                                 float* __restrict__ y) {
    __shared__ float sWt[64 * WT_STRIDE];   // 17.4 KB of 320 KB WGP LDS
    __shared__ float sB[64];
    int t = threadIdx.x;
    for (int i = 0; i < 16; ++i) {
        int idx = t + 256 * i;              // coalesced global read of Wc[k][n]
        int k = idx >> 6, n = idx & 63;
        sWt[n * WT_STRIDE + k] = Wc[idx];   // transpose into LDS
    }
    if (t < 64) sB[t] = bc[t];
    __syncthreads();

    int wave = t >> 5, lane = t & 31;
    int hi = lane >> 4;              // half-wave select
    int lo = lane & 15;
    int rowBase = blockIdx.x * 128 + wave * 16;

    int arow = rowBase + lo;                     // A-matrix row for this lane
    if (arow >= N_NODES) arow = N_NODES - 1;     // clamp tail: EXEC stays all-ones
    const float* xr = x + (long)arow * 64;

    const float* w0 = &sWt[(lo)      * WT_STRIDE];
    const float* w1 = &sWt[(16 + lo) * WT_STRIDE];
    const float* w2 = &sWt[(32 + lo) * WT_STRIDE];
    const float* w3 = &sWt[(48 + lo) * WT_STRIDE];

    v8f c0 = {}, c1 = {}, c2 = {}, c3 = {};
    for (int kk = 0; kk < 64; kk += 4) {
        int ka = kk + 2 * hi;                    // A/B layout: K = j + 2*(lane>>4)
        v2f a  = *(const v2f*)(xr + ka);         // one global_load_b64
        v2f b0 = *(const v2f*)(w0 + ka);         // one ds_load_b64 each, 8B aligned
        v2f b1 = *(const v2f*)(w1 + ka);
        v2f b2 = *(const v2f*)(w2 + ka);
        v2f b3 = *(const v2f*)(w3 + ka);
        c0 = __builtin_amdgcn_wmma_f32_16x16x4_f32(false, a, false, b0, (short)0, c0, false, false);
        c1 = __builtin_amdgcn_wmma_f32_16x16x4_f32(false, a, false, b1, (short)0, c1, false, false);
        c2 = __builtin_amdgcn_wmma_f32_16x16x4_f32(false, a, false, b2, (short)0, c2, false, false);
        c3 = __builtin_amdgcn_wmma_f32_16x16x4_f32(false, a, false, b3, (short)0, c3, false, false);
    }

    // D layout: reg i, lane L -> (M = i + 8*hi, N = nt*16 + lo)
    for (int i = 0; i < 8; ++i) {
        int row = rowBase + i + 8 * hi;
        if (row < N_NODES) {
            float* yr = y + (long)row * 64;
            yr[lo]      = c0[i] + sB[lo];
            yr[16 + lo] = c1[i] + sB[16 + lo];
            yr[32 + lo] = c2[i] + sB[32 + lo];
            yr[48 + lo] = c3[i] + sB[48 + lo];
        }
    }
}

// out[dst[e]][c] += y[src[e]][c]; 16 threads per edge, each handling 4 columns:
// one global_load_b128 gather + 4 f32 atomics (L2-resident: y and out are
// 25.6 MB each, well inside the 192 MB L2).
__global__ void scatter_edges(const int* __restrict__ ei, const float* __restrict__ y,
                              float* __restrict__ out) {
    long tid = (long)blockIdx.x * blockDim.x + threadIdx.x;
    long e = tid >> 4;
    int  g = (int)(tid & 15) << 2;
    if (e < N_EDGES) {
        int s = ei[e];             // edge_index[0][e] = src
        int d = ei[N_EDGES + e];   // edge_index[1][e] = dst
        float4 val = *(const float4*)(y + (long)s * 64 + g);
        float* o = out + (long)d * 64 + g;
        atomicAdd(o + 0, val.x);
        atomicAdd(o + 1, val.y);
        atomicAdd(o + 2, val.z);
        atomicAdd(o + 3, val.w);
    }
}

extern "C" void kernel_launch(void* const* d_in, const int* in_sizes, int n_in,
                              void* d_out, int out_size, void* d_ws, size_t ws_size,
                              hipStream_t stream) {
    // setup_inputs order: x, edge_index, Wq, bq, Wk, bk, Wv, bv, Wo, bo
    const float* x  = (const float*)d_in[0];
    const int*   ei = (const int*)d_in[1];
    const float* Wv = (const float*)d_in[6];
    const float* bv = (const float*)d_in[7];
    const float* Wo = (const float*)d_in[8];
    const float* bo = (const float*)d_in[9];
    float* out = (float*)d_out;

    float* ws = (float*)d_ws;
    float* Wc = ws;
    float* bc = ws + 4096;
    float* y  = ws + 8192;

    prep_weights<<<1, 256, 0, stream>>>(Wv, bv, Wo, bo, Wc, bc);

    int n4 = (N_NODES * 64) / 4;
    zero_out_kernel<<<(n4 + 255) / 256, 256, 0, stream>>>((float4*)out, n4);

    int gemmBlocks = (N_NODES + 127) / 128;   // 782 blocks x 8 waves
    gemm_wmma<<<gemmBlocks, 256, 0, stream>>>(x, Wc, bc, y);

    long nthr = (long)N_EDGES * 16;           // 16 threads per edge
    scatter_edges<<<(int)((nthr + 255) / 256), 256, 0, stream>>>(ei, y, out);
}